// GNNModel_38697655337346
// MI455X (gfx1250) — compile-verified
//
#include <hip/hip_runtime.h>

typedef __attribute__((ext_vector_type(2))) float v2f;
typedef __attribute__((ext_vector_type(8))) float v8f;

#define N_NODES_C 100000
#define FDIM 128

// ---------------- zero fill (float4, grid-stride) ----------------
__global__ void fill_zero_f4(float4* __restrict__ p, long long n4) {
  long long i = (long long)blockIdx.x * blockDim.x + threadIdx.x;
  long long stride = (long long)gridDim.x * blockDim.x;
  float4 z = make_float4(0.f, 0.f, 0.f, 0.f);
  for (; i < n4; i += stride) p[i] = z;
}

// ---------------- in-degree ----------------
__global__ void degree_kernel(const long long* __restrict__ dst,
                              float* __restrict__ deg, long long E) {
  long long e = (long long)blockIdx.x * blockDim.x + threadIdx.x;
  if (e < E) unsafeAtomicAdd(&deg[dst[e]], 1.0f);   // native global_atomic_add_f32
}

__global__ void deginv_kernel(float* __restrict__ deg, int n) {
  int i = blockIdx.x * blockDim.x + threadIdx.x;
  if (i < n) deg[i] = 1.0f / fmaxf(deg[i], 1.0f);
}

// ---------------- edge scatter: agg[dst] += x[src] ----------------
// One thread per (edge, feature). Both x (51.2MB) and agg (51.2MB) fit in the
// 192MB L2, so this runs at L2 gather/atomic throughput, not HBM.
__global__ void scatter_kernel(const float* __restrict__ x,
                               const long long* __restrict__ src,
                               const long long* __restrict__ dst,
                               float* __restrict__ agg, long long total) {
  long long gid = (long long)blockIdx.x * blockDim.x + threadIdx.x;
  if (gid >= total) return;
  long long e = gid >> 7;          // /128
  int f = (int)(gid & 127);
  long long s = src[e];
  long long d = dst[e];
  unsafeAtomicAdd(&agg[d * FDIM + f], x[s * FDIM + f]);
}

// ---------------- fused SAGE layer GEMM ----------------
// out = act( (agg .* dinv) @ wl^T + bias + x @ wr^T )
// One wave32 per 16x16 output tile, K stepped by 4 via V_WMMA_F32_16X16X4_F32.
// A layout (ISA 7.12.2, 32-bit A 16x4): lanes 0-15 carry K+{0,1}, lanes 16-31
// carry K+{2,3} for row M=lane&15  -> single v2f load per lane per step.
// B layout mirrors (VGPR0 = rows K,K+2 ; VGPR1 = rows K+1,K+3), B[k][n]=w[n][k].
__global__ void sage_wmma(const float* __restrict__ agg,
                          const float* __restrict__ dinv,
                          const float* __restrict__ xin,
                          const float* __restrict__ wl,
                          const float* __restrict__ wr,
                          const float* __restrict__ bias,
                          float* __restrict__ out, int relu) {
  int wave = (blockIdx.x * blockDim.x + threadIdx.x) >> 5;
  int lane = threadIdx.x & 31;
  int mt = wave >> 3;              // 8 N-tiles (128/16)
  int nt = wave & 7;
  if (mt >= N_NODES_C / 16) return;      // wave-uniform: EXEC stays all-ones

  int r15  = lane & 15;
  int hi   = lane >> 4;            // upper half-wave handles K+2,K+3
  int koff = hi * 2;
  int node = mt * 16 + r15;        // A row this lane streams
  int wrow = nt * 16 + r15;        // weight row (output feature) this lane streams
  float s = dinv[node];

  const float* aRow = agg + (long long)node * FDIM;
  const float* xRow = xin + (long long)node * FDIM;
  const float* wlR  = wl + wrow * FDIM;
  const float* wrR  = wr + wrow * FDIM;

  v8f c = {};
#pragma unroll
  for (int k = 0; k < FDIM; k += 4) {          // agg(scaled) @ wl^T
    v2f a = *(const v2f*)(aRow + k + koff);
    a.x *= s; a.y *= s;
    v2f b = *(const v2f*)(wlR + k + koff);
    c = __builtin_amdgcn_wmma_f32_16x16x4_f32(false, a, false, b,
                                              (short)0, c, false, false);
  }
#pragma unroll
  for (int k = 0; k < FDIM; k += 4) {          // + x @ wr^T (same accumulator)
    v2f a = *(const v2f*)(xRow + k + koff);
    v2f b = *(const v2f*)(wrR + k + koff);
    c = __builtin_amdgcn_wmma_f32_16x16x4_f32(false, a, false, b,
                                              (short)0, c, false, false);
  }

  // C/D layout: VGPR r -> (M = r + 8*hi, N = lane&15)
  int col   = nt * 16 + r15;
  float bv  = bias[col];
  int mbase = mt * 16 + hi * 8;
#pragma unroll
  for (int r = 0; r < 8; ++r) {
    float v = c[r] + bv;
    if (relu) v = fmaxf(v, 0.0f);
    out[(long long)(mbase + r) * FDIM + col] = v;   // coalesced across lanes
  }
}

// ---------------- final FC 128 -> 2, one wave per node ----------------
__global__ void fc_kernel(const float* __restrict__ h,
                          const float* __restrict__ wfc,
                          const float* __restrict__ bfc,
                          float* __restrict__ out) {
  int wave = (blockIdx.x * blockDim.x + threadIdx.x) >> 5;
  int lane = threadIdx.x & 31;
  if (wave >= N_NODES_C) return;
  float4 v  = ((const float4*)(h + (long long)wave * FDIM))[lane];
  float4 w0 = ((const float4*)(wfc))[lane];
  float4 w1 = ((const float4*)(wfc + FDIM))[lane];
  float p0 = v.x * w0.x + v.y * w0.y + v.z * w0.z + v.w * w0.w;
  float p1 = v.x * w1.x + v.y * w1.y + v.z * w1.z + v.w * w1.w;
#pragma unroll
  for (int o = 16; o > 0; o >>= 1) {
    p0 += __shfl_xor(p0, o, 32);
    p1 += __shfl_xor(p1, o, 32);
  }
  if (lane == 0) {
    out[wave * 2 + 0] = p0 + bfc[0];
    out[wave * 2 + 1] = p1 + bfc[1];
  }
}

extern "C" void kernel_launch(void* const* d_in, const int* in_sizes, int n_in,
                              void* d_out, int out_size, void* d_ws, size_t ws_size,
                              hipStream_t stream) {
  const float*     x   = (const float*)d_in[0];
  const long long* ei  = (const long long*)d_in[1];   // int64 edge_index [2,E]
  const float* w1l = (const float*)d_in[2];
  const float* b1  = (const float*)d_in[3];
  const float* w1r = (const float*)d_in[4];
  const float* w2l = (const float*)d_in[5];
  const float* b2  = (const float*)d_in[6];
  const float* w2r = (const float*)d_in[7];
  const float* w3l = (const float*)d_in[8];
  const float* b3  = (const float*)d_in[9];
  const float* w3r = (const float*)d_in[10];
  const float* wfc = (const float*)d_in[11];
  const float* bfc = (const float*)d_in[12];
  float* out = (float*)d_out;

  const long long E  = (long long)in_sizes[1] / 2;
  const long long N  = N_NODES_C;
  const long long NF = N * FDIM;
  const long long* src = ei;
  const long long* dst = ei + E;

  // workspace carve-out (256B aligned): deg | agg | h1 | h2
  char* ws = (char*)d_ws;
  size_t o = 0;
  auto carve = [&](size_t bytes) -> float* {
    float* p = (float*)(ws + o);
    o = (o + bytes + 255) & ~(size_t)255;
    return p;
  };
  float* deg = carve(N * 4);
  float* agg = carve(NF * 4);
  float* h1  = carve(NF * 4);
  float* h2  = carve(NF * 4);

  const int B = 256;                                   // 8 wave32s per block
  const long long total = E * FDIM;
  const int gemmBlocks  = (int)(N / 16);               // 8 waves/block, 8 N-tiles

  // degrees (depend only on dst -> computed once, inverted in place)
  fill_zero_f4<<<(int)((N / 4 + B - 1) / B), B, 0, stream>>>((float4*)deg, N / 4);
  degree_kernel<<<(int)((E + B - 1) / B), B, 0, stream>>>(dst, deg, E);
  deginv_kernel<<<(int)((N + B - 1) / B), B, 0, stream>>>(deg, (int)N);

  // layer 1: x -> h1
  fill_zero_f4<<<(int)((NF / 4 + B - 1) / B), B, 0, stream>>>((float4*)agg, NF / 4);
  scatter_kernel<<<(int)((total + B - 1) / B), B, 0, stream>>>(x, src, dst, agg, total);
  sage_wmma<<<gemmBlocks, B, 0, stream>>>(agg, deg, x, w1l, w1r, b1, h1, 1);

  // layer 2: h1 -> h2
  fill_zero_f4<<<(int)((NF / 4 + B - 1) / B), B, 0, stream>>>((float4*)agg, NF / 4);
  scatter_kernel<<<(int)((total + B - 1) / B), B, 0, stream>>>(h1, src, dst, agg, total);
  sage_wmma<<<gemmBlocks, B, 0, stream>>>(agg, deg, h1, w2l, w2r, b2, h2, 1);

  // layer 3: h2 -> h1 (ping-pong)
  fill_zero_f4<<<(int)((NF / 4 + B - 1) / B), B, 0, stream>>>((float4*)agg, NF / 4);
  scatter_kernel<<<(int)((total + B - 1) / B), B, 0, stream>>>(h2, src, dst, agg, total);
  sage_wmma<<<gemmBlocks, B, 0, stream>>>(agg, deg, h2, w3l, w3r, b3, h1, 1);

  // final FC: h1 -> out [N,2]
  fc_kernel<<<(int)(((long long)N * 32 + B - 1) / B), B, 0, stream>>>(h1, wfc, bfc, out);
}